// Detect_84765474554206
// MI455X (gfx1250) — compile-verified
//
#include <hip/hip_runtime.h>
#include <hip/hip_bf16.h>
#include <math.h>

// ---------------------------------------------------------------------------
// SSD Detect for MI455X (gfx1250, wave32).
//   B=128 images, A=8732 anchors, C=21 classes, K=200, conf_th=0.01, nms=0.45
// Pipeline:
//   K1: per-(b,a) softmax stats (rowmax, 1/sum(exp)) -> d_ws   (8.9 MB)
//   K2: one 256-thread block per (b, c in 1..20):
//         build order-encoded score keys in LDS (ties -> lower index, like
//         jax.lax.top_k), extract top-200 by incremental block argmax
//         (register-cached per-thread chunk maxima + wave32 shfl_xor trees),
//         async-gather the 200 loc/dbox rows into LDS via
//         global_load_async_to_lds_b128 / s_wait_asynccnt, decode boxes,
//         greedy NMS (1 barrier / iter), stable compaction, write (200,5).
//   K3: zero class 0 (reference does out.at[:,0].set(0)).
// ---------------------------------------------------------------------------

#define N_B 128
#define N_A 8732
#define N_C 21
#define N_K 200
#define CONF_TH 0.01f
#define NMS_TH 0.45f
#define ENC_NEGINF 0x007FFFFFu   // order-encoding of -inf (bits 0xFF800000 -> ~u)

__device__ __forceinline__ unsigned lds_off32(const void* p) {
  // generic shared address: low 32 bits are the LDS byte offset
  return (unsigned)(unsigned long long)p;
}

// --------------------------- Kernel 1: softmax stats ------------------------
__global__ __launch_bounds__(256) void softmax_stats_kernel(
    const float* __restrict__ conf, float2* __restrict__ stats, int nrows) {
  int row = blockIdx.x * blockDim.x + threadIdx.x;
  if (row >= nrows) return;
  const float* p = conf + (size_t)row * N_C;
  float m = p[0];
#pragma unroll
  for (int c = 1; c < N_C; ++c) m = fmaxf(m, p[c]);
  float s = 0.0f;
#pragma unroll
  for (int c = 0; c < N_C; ++c) s += expf(p[c] - m);
  stats[row] = make_float2(m, 1.0f / s);
}

// --------------------------- Kernel 2: topk + NMS ---------------------------
__global__ __launch_bounds__(256) void topk_nms_kernel(
    const float* __restrict__ loc, const float* __restrict__ conf,
    const float* __restrict__ dbox, const float2* __restrict__ stats,
    float* __restrict__ out) {
  const int tid = threadIdx.x;
  const int b   = blockIdx.x / (N_C - 1);
  const int c   = 1 + blockIdx.x % (N_C - 1);   // class 0 handled by K3

  __shared__ unsigned sh_enc[N_A];      // order-encoded masked scores (0 = taken)
  __shared__ float    stg[N_K * 8];     // async staging: loc[4] + dbox[4] per rank
  __shared__ float    tval[N_K];
  __shared__ int      tidxs[N_K];
  __shared__ float    bx1[N_K], by1[N_K], bx2[N_K], by2[N_K], bar_[N_K];
  __shared__ int      sup[N_K];
  __shared__ int      klist[N_K];
  __shared__ unsigned wred_e[8];
  __shared__ int      wred_i[8];
  __shared__ unsigned win_e;
  __shared__ int      win_i;
  __shared__ int      sh_nk;

  // ---- build keys; each thread privately owns anchors {tid, tid+256, ...}
  unsigned be = 0u; int bi = 0x7FFFFFFF;         // register-cached chunk argmax
  for (int a = tid; a < N_A; a += 256) {
    float2 st = stats[(size_t)b * N_A + a];
    float  p  = expf(conf[((size_t)b * N_A + a) * N_C + c] - st.x) * st.y;
    unsigned e = (p > CONF_TH) ? (__float_as_uint(p) | 0x80000000u) : ENC_NEGINF;
    sh_enc[a] = e;
    if (e > be || (e == be && a < bi)) { be = e; bi = a; }
  }

  // ---- 200 extraction rounds (incremental argmax)
  for (int r = 0; r < N_K; ++r) {
    unsigned re = be; int ri = bi;
#pragma unroll
    for (int m = 16; m >= 1; m >>= 1) {          // wave32 argmax tree
      unsigned oe = __shfl_xor(re, m, 32);
      int      oi = __shfl_xor(ri, m, 32);
      if (oe > re || (oe == re && oi < ri)) { re = oe; ri = oi; }
    }
    if ((tid & 31) == 0) { wred_e[tid >> 5] = re; wred_i[tid >> 5] = ri; }
    __syncthreads();
    if (tid == 0) {
      unsigned we = wred_e[0]; int wi = wred_i[0];
#pragma unroll
      for (int w = 1; w < 8; ++w) {
        unsigned oe = wred_e[w]; int oi = wred_i[w];
        if (oe > we || (oe == we && oi < wi)) { we = oe; wi = oi; }
      }
      win_e = we; win_i = wi;
      tval[r]  = (we == ENC_NEGINF) ? -INFINITY
                                    : __uint_as_float(we ^ 0x80000000u);
      tidxs[r] = wi;
    }
    __syncthreads();
    int wi = win_i;
    if ((wi & 255) == tid) {                     // only the owner rescans
      sh_enc[wi] = 0u;
      be = 0u; bi = 0x7FFFFFFF;
      for (int a = tid; a < N_A; a += 256) {
        unsigned e = sh_enc[a];
        if (e > be || (e == be && a < bi)) { be = e; bi = a; }
      }
    }
  }

  // ---- async gather of the 200 selected loc/dbox rows into LDS (CDNA5 path)
  if (tid < N_K) {
    int a = tidxs[tid];
    const float* lp = loc  + ((size_t)b * N_A + a) * 4;
    const float* dp = dbox + (size_t)a * 4;
    unsigned loff = lds_off32(&stg[tid * 8]);
    unsigned doff = lds_off32(&stg[tid * 8 + 4]);
    asm volatile("global_load_async_to_lds_b128 %0, %1, off"
                 :: "v"(loff), "v"(lp) : "memory");
    asm volatile("global_load_async_to_lds_b128 %0, %1, off"
                 :: "v"(doff), "v"(dp) : "memory");
  }
  asm volatile("s_wait_asynccnt 0x0" ::: "memory");
  __syncthreads();

  // ---- decode boxes
  if (tid < N_K) {
    float lx = stg[tid*8+0], ly = stg[tid*8+1], lw = stg[tid*8+2], lh = stg[tid*8+3];
    float dx = stg[tid*8+4], dy = stg[tid*8+5], dw = stg[tid*8+6], dh = stg[tid*8+7];
    float cx = dx * (1.0f + 0.1f * lx);
    float cy = dy * (1.0f + 0.1f * ly);
    float w  = dw * expf(0.2f * lw);
    float h  = dh * expf(0.2f * lh);
    float x1 = cx - w * 0.5f;
    float y1 = cy - h * 0.5f;
    float x2 = x1 + w;
    float y2 = y1 + h;
    bx1[tid] = x1; by1[tid] = y1; bx2[tid] = x2; by2[tid] = y2;
    bar_[tid] = (x2 - x1) * (y2 - y1);
    sup[tid] = 0;
  }
  if (tid == 0) sh_nk = 0;
  __syncthreads();

  // ---- greedy NMS (equivalent to the reference lax.scan: writes to
  //      suppressed[j<=i] in the reference are never re-read)
  int nk = 0;                                    // meaningful on thread 0 only
  for (int i = 0; i < N_K; ++i) {
    bool keep_i = (tval[i] != -INFINITY) && (sup[i] == 0);
    if (keep_i) {
      if (tid < N_K && tid > i && sup[tid] == 0) {
        float ix1 = fmaxf(bx1[i], bx1[tid]);
        float iy1 = fmaxf(by1[i], by1[tid]);
        float ix2 = fminf(bx2[i], bx2[tid]);
        float iy2 = fminf(by2[i], by2[tid]);
        float iw  = fmaxf(ix2 - ix1, 0.0f);
        float ih  = fmaxf(iy2 - iy1, 0.0f);
        float inter = iw * ih;
        float uni   = bar_[i] + bar_[tid] - inter;
        float iou   = inter / uni;               // 0/0 -> NaN -> no suppress (matches)
        if (iou > NMS_TH) sup[tid] = 1;
      }
      if (tid == 0) { klist[nk] = i; ++nk; }
    }
    __syncthreads();
  }
  if (tid == 0) sh_nk = nk;
  __syncthreads();

  // ---- stable compaction + store (kept first in score order, rest zero)
  if (tid < N_K) {
    float o0 = 0.f, o1 = 0.f, o2 = 0.f, o3 = 0.f, o4 = 0.f;
    if (tid < sh_nk) {
      int i = klist[tid];
      o0 = tval[i]; o1 = bx1[i]; o2 = by1[i]; o3 = bx2[i]; o4 = by2[i];
    }
    size_t base = (((size_t)b * N_C + c) * N_K + tid) * 5;
    out[base + 0] = o0; out[base + 1] = o1; out[base + 2] = o2;
    out[base + 3] = o3; out[base + 4] = o4;
  }
}

// --------------------------- Kernel 3: zero class 0 -------------------------
__global__ __launch_bounds__(256) void zero_class0_kernel(float* __restrict__ out) {
  int t = blockIdx.x * blockDim.x + threadIdx.x;   // N_B * N_K * 5 elements
  if (t < N_B * N_K * 5) {
    int b = t / (N_K * 5);
    int r = t % (N_K * 5);
    out[(size_t)b * N_C * N_K * 5 + r] = 0.0f;
  }
}

// ---------------------------------------------------------------------------
extern "C" void kernel_launch(void* const* d_in, const int* in_sizes, int n_in,
                              void* d_out, int out_size, void* d_ws, size_t ws_size,
                              hipStream_t stream) {
  (void)in_sizes; (void)n_in; (void)out_size; (void)ws_size;
  const float* loc  = (const float*)d_in[0];   // (B, A, 4)
  const float* conf = (const float*)d_in[1];   // (B, A, C)
  const float* dbox = (const float*)d_in[2];   // (A, 4)
  float*  out   = (float*)d_out;               // (B, C, 200, 5)
  float2* stats = (float2*)d_ws;               // (B*A) x {rowmax, 1/sumexp}

  const int rows = N_B * N_A;
  softmax_stats_kernel<<<(rows + 255) / 256, 256, 0, stream>>>(conf, stats, rows);
  topk_nms_kernel<<<N_B * (N_C - 1), 256, 0, stream>>>(loc, conf, dbox, stats, out);
  const int z = N_B * N_K * 5;
  zero_class0_kernel<<<(z + 255) / 256, 256, 0, stream>>>(out);
}